// GAT_32830730011175
// MI455X (gfx1250) — compile-verified
//
#include <hip/hip_runtime.h>
#include <hip/hip_bf16.h>

// ---------------------------------------------------------------------------
// GAT forward on MI455X (gfx1250, wave32).
// GEMMs use V_WMMA_F32_16X16X4_F32 (fp32 matrix pipe, exact fp32 numerics).
// W slab staged in LDS pre-paired as float2 so every B operand is a single
// aligned ds_load_b64 (no register re-pairing before WMMA).
// Edge softmax/aggregation via float atomics (node tables are L2-resident).
// ---------------------------------------------------------------------------

#define N_NODES 50000
#define N_EDGES 800000
#define N_GRAPHS 50
#define FEAT 256                 // IN_C == HEADS*HID_C == OUT_C == 256
#define ENC_NEG_INF 0x007FFFFFu  // enc(-inf)

typedef __attribute__((ext_vector_type(2))) float v2f;
typedef __attribute__((ext_vector_type(8))) float v8f;

// Monotonic float<->uint encoding so unsigned atomicMax == float max.
__device__ __forceinline__ unsigned enc_f(float f) {
    unsigned u = __float_as_uint(f);
    return (u & 0x80000000u) ? ~u : (u | 0x80000000u);
}
__device__ __forceinline__ float dec_f(unsigned x) {
    return (x & 0x80000000u) ? __uint_as_float(x & 0x7FFFFFFFu)
                             : __uint_as_float(~x);
}

// ---------------------------------------------------------------------------
__global__ void fill_u32(unsigned* __restrict__ p, unsigned v, int n) {
    int i = blockIdx.x * blockDim.x + threadIdx.x;
    if (i < n) p[i] = v;
}

// ---------------------------------------------------------------------------
// Y[M,256] = X[M,256] @ W[256,256] via V_WMMA_F32_16X16X4_F32.
//
// Block: 256 threads = 8 wave32s. Each block owns a 128-column N-slab:
// it stages W[:, n0:n0+128] into dynamic LDS as float2 pairs
//   lw2[p*128 + col] = { W[2p][n0+col], W[2p+1][n0+col] },  p = k/2
// (131072 B of the 320 KB WGP LDS). Each wave computes a 16x128 strip of Y
// with 8 f32 accumulators, so one global_load_b64 of A feeds 8 WMMAs.
//
// f32 16x4 A layout (ISA 7.12.2): lanes 0-15 -> M rows, VGPR0/1 = K {0,1}
// (lo half) / K {2,3} (hi half). B mirrors: lane needs pair p = k/2 + half.
// C/D: VGPR r = row r (lo half) / row r+8 (hi half).
// ---------------------------------------------------------------------------
__global__ __launch_bounds__(256) void gemm256_wmma_f32(
    const float* __restrict__ X, const float* __restrict__ W,
    float* __restrict__ Y, int Mtiles) {
    extern __shared__ v2f lw2[];  // 128 pairs (K) x 128 cols = 128 KB

    const int n0 = blockIdx.y * 128;
    for (int i = threadIdx.x; i < 128 * 128; i += 256) {
        int p = i >> 7, col = i & 127;
        v2f w;
        w.x = W[(2 * p) * 256 + n0 + col];
        w.y = W[(2 * p + 1) * 256 + n0 + col];
        lw2[i] = w;  // ds_store_b64
    }
    __syncthreads();

    const int wave  = threadIdx.x >> 5;
    const int mtile = blockIdx.x * 8 + wave;
    if (mtile >= Mtiles) return;  // wave-uniform: EXEC stays all-1s for WMMA

    const int lane = threadIdx.x & 31;
    const int l16  = lane & 15;
    const int half = lane >> 4;  // 0: K+{0,1}, 1: K+{2,3}

    const float* Xrow = X + (size_t)(mtile * 16 + l16) * FEAT;

    v8f a0 = {}, a1 = {}, a2 = {}, a3 = {}, a4 = {}, a5 = {}, a6 = {}, a7 = {};

#pragma unroll 2
    for (int k = 0; k < 256; k += 4) {
        v2f a = *(const v2f*)(Xrow + k + 2 * half);  // global_load_b64
        const v2f* bp = lw2 + ((k >> 1) + half) * 128 + l16;
        v2f b0 = bp[0];    // ds_load_b64 (pairs pre-packed in LDS)
        v2f b1 = bp[16];
        v2f b2 = bp[32];
        v2f b3 = bp[48];
        v2f b4 = bp[64];
        v2f b5 = bp[80];
        v2f b6 = bp[96];
        v2f b7 = bp[112];
        a0 = __builtin_amdgcn_wmma_f32_16x16x4_f32(false, a, false, b0, (short)0, a0, false, false);
        a1 = __builtin_amdgcn_wmma_f32_16x16x4_f32(false, a, false, b1, (short)0, a1, false, false);
        a2 = __builtin_amdgcn_wmma_f32_16x16x4_f32(false, a, false, b2, (short)0, a2, false, false);
        a3 = __builtin_amdgcn_wmma_f32_16x16x4_f32(false, a, false, b3, (short)0, a3, false, false);
        a4 = __builtin_amdgcn_wmma_f32_16x16x4_f32(false, a, false, b4, (short)0, a4, false, false);
        a5 = __builtin_amdgcn_wmma_f32_16x16x4_f32(false, a, false, b5, (short)0, a5, false, false);
        a6 = __builtin_amdgcn_wmma_f32_16x16x4_f32(false, a, false, b6, (short)0, a6, false, false);
        a7 = __builtin_amdgcn_wmma_f32_16x16x4_f32(false, a, false, b7, (short)0, a7, false, false);
    }

    const int rbase = mtile * 16 + half * 8;  // rows r (lo) / r+8 (hi)
#pragma unroll
    for (int r = 0; r < 8; r++) {
        size_t off = (size_t)(rbase + r) * FEAT + n0 + l16;
        Y[off]       = a0[r];
        Y[off + 16]  = a1[r];
        Y[off + 32]  = a2[r];
        Y[off + 48]  = a3[r];
        Y[off + 64]  = a4[r];
        Y[off + 80]  = a5[r];
        Y[off + 96]  = a6[r];
        Y[off + 112] = a7[r];
    }
}

// ---------------------------------------------------------------------------
// a_src[n,h] = sum_c h[n,h,c]*att_src[h,c]; same for a_dst. Thread per (n,h).
// ---------------------------------------------------------------------------
__global__ void att_reduce(const float* __restrict__ h,
                           const float* __restrict__ att_s,
                           const float* __restrict__ att_d,
                           float* __restrict__ out_s, float* __restrict__ out_d,
                           int N, int H, int C) {
    int gid = blockIdx.x * blockDim.x + threadIdx.x;
    if (gid >= N * H) return;
    int n = gid / H, hh = gid % H;
    const float* row = h + (size_t)n * H * C + hh * C;
    const float* as = att_s + hh * C;
    const float* ad = att_d + hh * C;
    float sa = 0.f, sd = 0.f;
    for (int c = 0; c < C; c++) {
        float v = row[c];
        sa += v * as[c];
        sd += v * ad[c];
    }
    out_s[gid] = sa;
    out_d[gid] = sd;
}

// ---------------------------------------------------------------------------
// Edge pass 1: segment max of leaky_relu(a_src[src]+a_dst[dst]) over dst.
// ---------------------------------------------------------------------------
__global__ void edge_max(const float* __restrict__ asrc,
                         const float* __restrict__ adst,
                         const int* __restrict__ src, const int* __restrict__ dst,
                         unsigned* __restrict__ m, int E, int H) {
    int gid = blockIdx.x * blockDim.x + threadIdx.x;
    if (gid >= E * H) return;
    int e = gid / H, hh = gid % H;
    int s = src[e], d = dst[e];
    float v = asrc[s * H + hh] + adst[d * H + hh];
    v = (v > 0.f) ? v : 0.2f * v;  // NEG_SLOPE = 0.2
    atomicMax(&m[d * H + hh], enc_f(v));
}

// ---------------------------------------------------------------------------
// Edge pass 2: ex = exp(e - m[dst]); segment-sum into s; stash ex.
// ---------------------------------------------------------------------------
__global__ void edge_expsum(const float* __restrict__ asrc,
                            const float* __restrict__ adst,
                            const int* __restrict__ src,
                            const int* __restrict__ dst,
                            const unsigned* __restrict__ m,
                            float* __restrict__ ssum, float* __restrict__ ex,
                            int E, int H) {
    int gid = blockIdx.x * blockDim.x + threadIdx.x;
    if (gid >= E * H) return;
    int e = gid / H, hh = gid % H;
    int s = src[e], d = dst[e];
    float v = asrc[s * H + hh] + adst[d * H + hh];
    v = (v > 0.f) ? v : 0.2f * v;
    float exv = __expf(v - dec_f(m[d * H + hh]));
    atomicAdd(&ssum[d * H + hh], exv);
    ex[gid] = exv;
}

// ---------------------------------------------------------------------------
// Edge pass 3: out[dst] += h[src]*alpha. One block per edge, 256 lanes wide
// (coalesced 1 KB gather from L2-resident h; float atomics to L2).
// ---------------------------------------------------------------------------
__global__ __launch_bounds__(256) void edge_aggregate(
    const float* __restrict__ hfeat, const int* __restrict__ src,
    const int* __restrict__ dst, const float* __restrict__ ex,
    const float* __restrict__ ssum, float* __restrict__ out, int H, int C) {
    int e = blockIdx.x;
    int t = threadIdx.x;  // 0..255 == H*C
    int hh = t / C;
    int s = src[e], d = dst[e];
    float alpha = ex[(size_t)e * H + hh] / ssum[d * H + hh];
    atomicAdd(&out[(size_t)d * FEAT + t], hfeat[(size_t)s * FEAT + t] * alpha);
}

// ---------------------------------------------------------------------------
__global__ void bias_relu(float* __restrict__ x, const float* __restrict__ b,
                          int total) {
    int i = blockIdx.x * blockDim.x + threadIdx.x;
    if (i < total) x[i] = fmaxf(x[i] + b[i & (FEAT - 1)], 0.f);
}

__global__ void count_nodes(const int* __restrict__ batch,
                            float* __restrict__ cnt, int N) {
    int n = blockIdx.x * blockDim.x + threadIdx.x;
    if (n < N) atomicAdd(&cnt[batch[n]], 1.0f);
}

__global__ void pool_sum(const float* __restrict__ x,
                         const int* __restrict__ batch,
                         float* __restrict__ pooled, int N) {
    int i = blockIdx.x * blockDim.x + threadIdx.x;
    if (i >= N * FEAT) return;
    int n = i >> 8, c = i & (FEAT - 1);
    atomicAdd(&pooled[batch[n] * FEAT + c], x[i]);
}

__global__ __launch_bounds__(256) void fc_relu(
    const float* __restrict__ pooled, const float* __restrict__ cnt,
    const float* __restrict__ Wfc, const float* __restrict__ bfc,
    float* __restrict__ out) {
    int g = blockIdx.x, c = threadIdx.x;
    float inv = 1.0f / fmaxf(cnt[g], 1.0f);
    float s = 0.f;
    for (int k = 0; k < FEAT; k++)
        s += pooled[g * FEAT + k] * inv * Wfc[k * FEAT + c];
    out[g * FEAT + c] = fmaxf(s + bfc[c], 0.f);
}

// ---------------------------------------------------------------------------
extern "C" void kernel_launch(void* const* d_in, const int* in_sizes, int n_in,
                              void* d_out, int out_size, void* d_ws,
                              size_t ws_size, hipStream_t stream) {
    const float* x        = (const float*)d_in[0];
    const int*   eidx     = (const int*)d_in[1];
    const int*   batch    = (const int*)d_in[2];
    const float* W1       = (const float*)d_in[3];
    const float* att_src1 = (const float*)d_in[4];
    const float* att_dst1 = (const float*)d_in[5];
    const float* b1       = (const float*)d_in[6];
    const float* W2       = (const float*)d_in[7];
    const float* att_src2 = (const float*)d_in[8];
    const float* att_dst2 = (const float*)d_in[9];
    const float* b2       = (const float*)d_in[10];
    const float* Wfc      = (const float*)d_in[11];
    const float* bfc      = (const float*)d_in[12];
    float* out = (float*)d_out;

    const int* src = eidx;            // edge_index[0]
    const int* dst = eidx + N_EDGES;  // edge_index[1]

    // ---- workspace carve-up (floats) ------------------------------------
    const size_t NC = (size_t)N_NODES * FEAT;  // 12.8M
    float* p = (float*)d_ws;
    float*    bufA   = p; p += NC;             // h1, then h2
    float*    bufB   = p; p += NC;             // out1/x2, then out2
    float*    asrc1  = p; p += (size_t)N_NODES * 4;
    float*    adst1  = p; p += (size_t)N_NODES * 4;
    unsigned* m1     = (unsigned*)p; p += (size_t)N_NODES * 4;
    float*    s1     = p; p += (size_t)N_NODES * 4;
    float*    ex1    = p; p += (size_t)N_EDGES * 4;
    float*    asrc2  = p; p += N_NODES;
    float*    adst2  = p; p += N_NODES;
    unsigned* m2     = (unsigned*)p; p += N_NODES;
    float*    s2     = p; p += N_NODES;
    float*    ex2    = p; p += N_EDGES;
    float*    pooled = p; p += (size_t)N_GRAPHS * FEAT;
    float*    cnt    = p; p += N_GRAPHS;

    const int Mtiles = N_NODES / 16;            // 3125 exactly
    const size_t lds_bytes = 128 * 128 * sizeof(v2f);  // 128 KB paired W slab
    dim3 ggrid((Mtiles + 7) / 8, 2);            // 391 x 2 (128-col N-slabs)

    auto cdiv = [](long long a, long long b) { return (int)((a + b - 1) / b); };

    // =========================== Layer 1 (H=4, C=64) =====================
    gemm256_wmma_f32<<<ggrid, 256, lds_bytes, stream>>>(x, W1, bufA, Mtiles);
    att_reduce<<<cdiv(N_NODES * 4, 256), 256, 0, stream>>>(
        bufA, att_src1, att_dst1, asrc1, adst1, N_NODES, 4, 64);

    fill_u32<<<cdiv(N_NODES * 4, 256), 256, 0, stream>>>(m1, ENC_NEG_INF,
                                                         N_NODES * 4);
    fill_u32<<<cdiv(N_NODES * 4, 256), 256, 0, stream>>>((unsigned*)s1, 0u,
                                                         N_NODES * 4);
    fill_u32<<<cdiv((long long)NC, 256), 256, 0, stream>>>((unsigned*)bufB, 0u,
                                                           (int)NC);

    edge_max<<<cdiv((long long)N_EDGES * 4, 256), 256, 0, stream>>>(
        asrc1, adst1, src, dst, m1, N_EDGES, 4);
    edge_expsum<<<cdiv((long long)N_EDGES * 4, 256), 256, 0, stream>>>(
        asrc1, adst1, src, dst, m1, s1, ex1, N_EDGES, 4);
    edge_aggregate<<<N_EDGES, 256, 0, stream>>>(bufA, src, dst, ex1, s1, bufB,
                                                4, 64);
    bias_relu<<<cdiv((long long)NC, 256), 256, 0, stream>>>(bufB, b1, (int)NC);

    // =========================== Layer 2 (H=1, C=256) ====================
    gemm256_wmma_f32<<<ggrid, 256, lds_bytes, stream>>>(bufB, W2, bufA, Mtiles);
    att_reduce<<<cdiv(N_NODES, 256), 256, 0, stream>>>(
        bufA, att_src2, att_dst2, asrc2, adst2, N_NODES, 1, 256);

    fill_u32<<<cdiv(N_NODES, 256), 256, 0, stream>>>(m2, ENC_NEG_INF, N_NODES);
    fill_u32<<<cdiv(N_NODES, 256), 256, 0, stream>>>((unsigned*)s2, 0u, N_NODES);
    fill_u32<<<cdiv((long long)NC, 256), 256, 0, stream>>>((unsigned*)bufB, 0u,
                                                           (int)NC);

    edge_max<<<cdiv(N_EDGES, 256), 256, 0, stream>>>(asrc2, adst2, src, dst, m2,
                                                     N_EDGES, 1);
    edge_expsum<<<cdiv(N_EDGES, 256), 256, 0, stream>>>(asrc2, adst2, src, dst,
                                                        m2, s2, ex2, N_EDGES, 1);
    edge_aggregate<<<N_EDGES, 256, 0, stream>>>(bufA, src, dst, ex2, s2, bufB,
                                                1, 256);
    bias_relu<<<cdiv((long long)NC, 256), 256, 0, stream>>>(bufB, b2, (int)NC);

    // ======================= mean-pool + FC + ReLU =======================
    fill_u32<<<cdiv(N_GRAPHS * FEAT + N_GRAPHS, 256), 256, 0, stream>>>(
        (unsigned*)pooled, 0u, N_GRAPHS * FEAT + N_GRAPHS);  // pooled+cnt contig
    count_nodes<<<cdiv(N_NODES, 256), 256, 0, stream>>>(batch, cnt, N_NODES);
    pool_sum<<<cdiv((long long)NC, 256), 256, 0, stream>>>(bufB, batch, pooled,
                                                           N_NODES);
    fc_relu<<<N_GRAPHS, 256, 0, stream>>>(pooled, cnt, Wfc, bfc, out);
}